// MACEWrapper_10015863734371
// MI455X (gfx1250) — compile-verified
//
#include <hip/hip_runtime.h>
#include <math.h>

// ---------------- problem constants ----------------
#define NAT    4096
#define FEAT   128
#define NEL    4
#define NB     8
#define NCHUNK 16         // j-split for pairwise kernels (deterministic partials)
#define JCH    (NAT / NCHUNK)
#define JTILE  256

#define BOHR2ANG 0.529177210903f
#define EV2HA    0.0367493221756f
#define GSCALE   (0.0367493221756f * 0.529177210903f)   // eV/A -> Ha/Bohr on dE/dx
#define CUT      5.0f
#define BPI      0.6283185307179586f                     // pi / CUT
#define SQ2C     0.6324555320336759f                     // sqrt(2/CUT)

// workspace layout (float offsets)
#define OFF_POS   0u
#define OFF_Z     12288u
#define OFF_M     16384u
#define OFF_SPART 20480u       // [NCHUNK][NAT][32]
#define OFF_S     2117632u     // [NAT][32]
#define OFF_AGG   2248704u     // [NAT][128]
#define OFF_H     2772992u
#define OFF_T     3297280u
#define OFF_G     3821568u
#define OFF_BC    4345856u     // [NAT][32]
#define OFF_GPART 4476928u     // [NCHUNK][NAT][3]
#define OFF_EPART 4673536u     // [16]

typedef float v2f __attribute__((ext_vector_type(2)));
typedef float v8f __attribute__((ext_vector_type(8)));

// ---------------- prep: z lookup + unit conversion ----------------
__global__ void k_prep(const float* __restrict__ posb, const int* __restrict__ an,
                       const int* __restrict__ zt, float* __restrict__ pos,
                       int* __restrict__ zidx) {
  int i = blockIdx.x * blockDim.x + threadIdx.x;
  if (i >= NAT) return;
  int a = an[i];
  int t = 0;                      // argmax(first match); no-match -> 0 like jnp.argmax
  if      (a == zt[0]) t = 0;
  else if (a == zt[1]) t = 1;
  else if (a == zt[2]) t = 2;
  else if (a == zt[3]) t = 3;
  zidx[i] = t;
  pos[3*i+0] = posb[3*i+0] * BOHR2ANG;
  pos[3*i+1] = posb[3*i+1] * BOHR2ANG;
  pos[3*i+2] = posb[3*i+2] * BOHR2ANG;
}

// M[(t*8+k)*128 + f] = W_embed[t,f] * W_rad[k,f]
__global__ void k_make_m(const float* __restrict__ We, const float* __restrict__ Wr,
                         float* __restrict__ M) {
  int idx = blockIdx.x * blockDim.x + threadIdx.x;   // 32*128
  if (idx >= 32 * FEAT) return;
  int m = idx >> 7, f = idx & 127;
  int t = m >> 3, k = m & 7;
  M[idx] = We[t * FEAT + f] * Wr[k * FEAT + f];
}

// ---------------- pairwise forward: S_i[t,k] = sum_j rbf ----------------
__global__ __launch_bounds__(256) void k_pair_fwd(const float* __restrict__ pos,
                                                  const int* __restrict__ zidx,
                                                  float* __restrict__ Spart) {
  __shared__ float spx[JTILE], spy[JTILE], spz[JTILE];
  __shared__ int   sz[JTILE];
  const int tid = threadIdx.x;
  const int i = blockIdx.x * 256 + tid;
  const float px = pos[3*i], py = pos[3*i+1], pz = pos[3*i+2];
  float S[32];
#pragma unroll
  for (int m = 0; m < 32; ++m) S[m] = 0.f;

  const int jbase = blockIdx.y * JCH;
  for (int jt = 0; jt < JCH; jt += JTILE) {
    int j = jbase + jt + tid;
    spx[tid] = pos[3*j]; spy[tid] = pos[3*j+1]; spz[tid] = pos[3*j+2];
    sz[tid]  = zidx[j];
    __syncthreads();
    for (int jj = 0; jj < JTILE; ++jj) {
      float dx = px - spx[jj], dy = py - spy[jj], dz = pz - spz[jj];
      float r = sqrtf(dx*dx + dy*dy + dz*dz);
      if (r < CUT && r > 1e-8f) {
        float x = BPI * r, sx, cx;
        sincosf(x, &sx, &cx);
        float env  = 0.5f * (cx + 1.0f);
        float pref = SQ2C * env / r;
        float rb[NB];
        float sk = sx, ck = cx;
        rb[0] = pref * sk;
#pragma unroll
        for (int k = 1; k < NB; ++k) {                 // sin(kx) recurrence
          float sn = sk*cx + ck*sx, cn = ck*cx - sk*sx;
          sk = sn; ck = cn;
          rb[k] = pref * sk;
        }
        int tz = sz[jj];                               // wave-uniform -> scalar branch
#define ACC8(OFF) { S[OFF+0]+=rb[0]; S[OFF+1]+=rb[1]; S[OFF+2]+=rb[2]; S[OFF+3]+=rb[3]; \
                    S[OFF+4]+=rb[4]; S[OFF+5]+=rb[5]; S[OFF+6]+=rb[6]; S[OFF+7]+=rb[7]; }
        if      (tz == 0) ACC8(0)
        else if (tz == 1) ACC8(8)
        else if (tz == 2) ACC8(16)
        else              ACC8(24)
#undef ACC8
      }
    }
    __syncthreads();
  }
  float* out = Spart + ((size_t)blockIdx.y * NAT + i) * 32;
#pragma unroll
  for (int m = 0; m < 32; ++m) out[m] = S[m];
}

__global__ void k_reduce_s(const float* __restrict__ Spart, float* __restrict__ S) {
  int idx = blockIdx.x * blockDim.x + threadIdx.x;     // NAT*32
  if (idx >= NAT * 32) return;
  float a = 0.f;
#pragma unroll
  for (int c = 0; c < NCHUNK; ++c) a += Spart[(size_t)c * NAT * 32 + idx];
  S[idx] = a;
}

// ---------------- f32 WMMA GEMM (V_WMMA_F32_16X16X4_F32) ----------------
// One wave computes one 16x16 f32 tile of C = A[MxKD] * B (B is [KD x ND],
// optionally accessed transposed as B[n*KD+k]).  EPI==1: h=silu(u) -> C,
// t = w_out .* silu'(u) -> C2.
template <int KD, int ND, bool TB, int EPI>
__global__ __launch_bounds__(256) void k_wmma_gemm(const float* __restrict__ A,
                                                   const float* __restrict__ Bm,
                                                   float* __restrict__ C,
                                                   float* __restrict__ C2,
                                                   const float* __restrict__ wout) {
  const int lane = threadIdx.x & 31;
  const int wave = threadIdx.x >> 5;
  const int tile = blockIdx.x * (blockDim.x >> 5) + wave;
  const int ntc  = ND / 16;
  const int tr = tile / ntc, tc = tile - tr * ntc;
  const int row0 = tr * 16, col0 = tc * 16;
  const int half = lane >> 4, mn = lane & 15;

  v8f acc = {};
  for (int k0 = 0; k0 < KD; k0 += 4) {
    const int ka = k0 + (half ? 2 : 0);
    v2f a, b;
    a.x = A[(size_t)(row0 + mn) * KD + ka];
    a.y = A[(size_t)(row0 + mn) * KD + ka + 1];
    if (TB) {  // B[k][n] = Bm[n*KD + k]
      b.x = Bm[(size_t)(col0 + mn) * KD + ka];
      b.y = Bm[(size_t)(col0 + mn) * KD + ka + 1];
    } else {   // row-major [KD][ND]
      b.x = Bm[(size_t)ka * ND + col0 + mn];
      b.y = Bm[(size_t)(ka + 1) * ND + col0 + mn];
    }
    acc = __builtin_amdgcn_wmma_f32_16x16x4_f32(false, a, false, b,
                                                (short)0, acc, false, false);
  }
  const int mbase = half * 8;
#pragma unroll
  for (int v = 0; v < 8; ++v) {
    float d = acc[v];
    size_t o = (size_t)(row0 + mbase + v) * ND + col0 + mn;
    if (EPI == 1) {
      float sg = 1.0f / (1.0f + expf(-d));
      C[o]  = d * sg;                                      // h = silu(u)
      C2[o] = wout[col0 + mn] * sg * (1.0f + d * (1.0f - sg)); // w .* silu'(u)
    } else {
      C[o] = d;
    }
  }
}

// ---------------- energy reduction (deterministic) ----------------
__global__ void k_energy_partial(const float* __restrict__ h, const float* __restrict__ wout,
                                 const int* __restrict__ zidx, const float* __restrict__ ae,
                                 float* __restrict__ epart) {
  __shared__ float red[256];
  __shared__ float sw[FEAT];
  int tid = threadIdx.x;
  if (tid < FEAT) sw[tid] = wout[tid];
  __syncthreads();
  int i = blockIdx.x * 256 + tid;
  float e = ae[zidx[i]];
  const float* hp = h + (size_t)i * FEAT;
#pragma unroll 8
  for (int f = 0; f < FEAT; ++f) e += hp[f] * sw[f];
  red[tid] = e;
  __syncthreads();
  for (int s = 128; s >= 1; s >>= 1) {
    if (tid < s) red[tid] += red[tid + s];
    __syncthreads();
  }
  if (tid == 0) epart[blockIdx.x] = red[0];
}

__global__ void k_energy_final(const float* __restrict__ epart, float* __restrict__ out) {
  if (threadIdx.x == 0) {
    float s = 0.f;
    for (int c = 0; c < 16; ++c) s += epart[c];
    out[0] = s * EV2HA;
  }
}

// ---------------- pairwise backward ----------------
__global__ __launch_bounds__(256) void k_pair_bwd(const float* __restrict__ pos,
                                                  const int* __restrict__ zidx,
                                                  const float* __restrict__ Bc,
                                                  float* __restrict__ gpart) {
  __shared__ float spx[JTILE], spy[JTILE], spz[JTILE];
  __shared__ int   sz[JTILE];
  __shared__ float sB[JTILE * 32];
  const int tid = threadIdx.x;
  const int i = blockIdx.x * 256 + tid;
  const float px = pos[3*i], py = pos[3*i+1], pz = pos[3*i+2];
  const int basei = zidx[i] * 8;
  float Bi[32];
#pragma unroll
  for (int m = 0; m < 32; ++m) Bi[m] = Bc[(size_t)i * 32 + m];
  float gx = 0.f, gy = 0.f, gz = 0.f;

  const int jbase = blockIdx.y * JCH;
  for (int jt = 0; jt < JCH; jt += JTILE) {
    int j = jbase + jt + tid;
    spx[tid] = pos[3*j]; spy[tid] = pos[3*j+1]; spz[tid] = pos[3*j+2];
    sz[tid]  = zidx[j];
#pragma unroll
    for (int m = 0; m < 32; ++m) sB[tid * 32 + m] = Bc[(size_t)j * 32 + m];
    __syncthreads();
    for (int jj = 0; jj < JTILE; ++jj) {
      float dx = px - spx[jj], dy = py - spy[jj], dz = pz - spz[jj];
      float r = sqrtf(dx*dx + dy*dy + dz*dz);
      if (r < CUT && r > 1e-8f) {
        float rinv = 1.0f / r;
        float x = BPI * r, sx, cx;
        sincosf(x, &sx, &cx);
        float env = 0.5f * (cx + 1.0f);
        float t2  = (-0.5f * BPI) * sx - rinv * env;   // env' - env/r
        float pref = SQ2C * rinv;
        float fkp[NB];
        float sk = sx, ck = cx;
#pragma unroll
        for (int k = 0; k < NB; ++k) {                 // d/dr of rbf_k * env
          float ak = (float)(k + 1) * BPI;
          fkp[k] = pref * (ak * ck * env + sk * t2);
          float sn = sk*cx + ck*sx, cn = ck*cx - sk*sx;
          sk = sn; ck = cn;
        }
        // cB: B[j][z_i*8+k] (per-lane z_i) from LDS
        const float* Bj = sB + jj * 32 + basei;
        float coef = fkp[0]*Bj[0] + fkp[1]*Bj[1] + fkp[2]*Bj[2] + fkp[3]*Bj[3]
                   + fkp[4]*Bj[4] + fkp[5]*Bj[5] + fkp[6]*Bj[6] + fkp[7]*Bj[7];
        // cA: B[i][z_j*8+k], z_j wave-uniform -> scalar branch, static reg indices
        int tz = sz[jj];
#define DOT8(OFF) (fkp[0]*Bi[OFF+0]+fkp[1]*Bi[OFF+1]+fkp[2]*Bi[OFF+2]+fkp[3]*Bi[OFF+3]+ \
                   fkp[4]*Bi[OFF+4]+fkp[5]*Bi[OFF+5]+fkp[6]*Bi[OFF+6]+fkp[7]*Bi[OFF+7])
        if      (tz == 0) coef += DOT8(0);
        else if (tz == 1) coef += DOT8(8);
        else if (tz == 2) coef += DOT8(16);
        else              coef += DOT8(24);
#undef DOT8
        float w = coef * rinv;
        gx += w * dx; gy += w * dy; gz += w * dz;
      }
    }
    __syncthreads();
  }
  float* out = gpart + ((size_t)blockIdx.y * NAT + i) * 3;
  out[0] = gx; out[1] = gy; out[2] = gz;
}

__global__ void k_grad_final(const float* __restrict__ gpart, float* __restrict__ out) {
  int idx = blockIdx.x * blockDim.x + threadIdx.x;     // NAT*3
  if (idx >= NAT * 3) return;
  float g = 0.f;
#pragma unroll
  for (int c = 0; c < NCHUNK; ++c) g += gpart[(size_t)c * NAT * 3 + idx];
  out[1 + idx] = g * GSCALE;                           // dE/dpos -> Ha/Bohr gradient
}

// ---------------- launch ----------------
extern "C" void kernel_launch(void* const* d_in, const int* in_sizes, int n_in,
                              void* d_out, int out_size, void* d_ws, size_t ws_size,
                              hipStream_t stream) {
  (void)in_sizes; (void)n_in; (void)out_size; (void)ws_size;
  const float* posb = (const float*)d_in[0];
  const int*   an   = (const int*)d_in[1];
  const int*   zt   = (const int*)d_in[2];
  const float* We   = (const float*)d_in[3];
  const float* Wr   = (const float*)d_in[4];
  const float* W1   = (const float*)d_in[5];
  const float* wout = (const float*)d_in[6];
  const float* ae   = (const float*)d_in[7];
  float* out = (float*)d_out;

  float* ws    = (float*)d_ws;
  float* pos   = ws + OFF_POS;
  int*   zidx  = (int*)(ws + OFF_Z);
  float* M     = ws + OFF_M;
  float* Spart = ws + OFF_SPART;
  float* S     = ws + OFF_S;
  float* agg   = ws + OFF_AGG;
  float* h     = ws + OFF_H;
  float* tbuf  = ws + OFF_T;
  float* g     = ws + OFF_G;
  float* Bcb   = ws + OFF_BC;
  float* gpart = ws + OFF_GPART;
  float* epart = ws + OFF_EPART;

  k_prep<<<NAT/256, 256, 0, stream>>>(posb, an, zt, pos, zidx);
  k_make_m<<<(32*FEAT)/256, 256, 0, stream>>>(We, Wr, M);

  k_pair_fwd<<<dim3(NAT/256, NCHUNK), 256, 0, stream>>>(pos, zidx, Spart);
  k_reduce_s<<<(NAT*32)/256, 256, 0, stream>>>(Spart, S);

  // agg = S @ M                      [4096x32]x[32x128]
  k_wmma_gemm<32, 128, false, 0><<<(NAT/16)*(128/16)/8, 256, 0, stream>>>(S, M, agg, nullptr, nullptr);
  // u = agg @ W1; h = silu(u); t = w_out .* silu'(u)
  k_wmma_gemm<128, 128, false, 1><<<(NAT/16)*(128/16)/8, 256, 0, stream>>>(agg, W1, h, tbuf, wout);
  // g = t @ W1^T
  k_wmma_gemm<128, 128, true, 0><<<(NAT/16)*(128/16)/8, 256, 0, stream>>>(tbuf, W1, g, nullptr, nullptr);
  // B = g @ M^T                      [4096x128]x[128x32]
  k_wmma_gemm<128, 32, true, 0><<<(NAT/16)*(32/16)/8, 256, 0, stream>>>(g, M, Bcb, nullptr, nullptr);

  k_energy_partial<<<NAT/256, 256, 0, stream>>>(h, wout, zidx, ae, epart);
  k_energy_final<<<1, 32, 0, stream>>>(epart, out);

  k_pair_bwd<<<dim3(NAT/256, NCHUNK), 256, 0, stream>>>(pos, zidx, Bcb, gpart);
  k_grad_final<<<(NAT*3)/256, 256, 0, stream>>>(gpart, out);
}